// GCN_20804821581912
// MI455X (gfx1250) — compile-verified
//
#include <hip/hip_runtime.h>

#define EMB 128
#define WSTRIDE 132   // padded LDS row stride (floats): rows k,k+2 differ by 8 banks

typedef float v2f __attribute__((ext_vector_type(2)));
typedef float v8f __attribute__((ext_vector_type(8)));

// Native f32 atomic add (global_atomic_add_f32), avoids CAS-loop lowering.
__device__ __forceinline__ void atomic_fadd(float* p, float v) {
  unsafeAtomicAdd(p, v);
}

__global__ __launch_bounds__(256) void k_zero_i32(int* __restrict__ p, int n) {
  int i = blockIdx.x * 256 + threadIdx.x;
  if (i < n) p[i] = 0;
}

// In-degree over the concatenated edge list (same graph both layers).
__global__ __launch_bounds__(256) void k_degree(const int* __restrict__ dstA,
                                                const int* __restrict__ dstB,
                                                int E1, int totalE,
                                                int* __restrict__ deg) {
  int e = blockIdx.x * 256 + threadIdx.x;
  if (e >= totalE) return;
  int d = (e < E1) ? dstA[e] : dstB[e - E1];
  atomicAdd(&deg[d], 1);
}

__global__ __launch_bounds__(256) void k_dinv(const int* __restrict__ deg,
                                              float* __restrict__ dinv, int n) {
  int i = blockIdx.x * 256 + threadIdx.x;
  if (i < n) dinv[i] = rsqrtf((float)deg[i] + 1.0f);  // +1 self-loop
}

// H = X * W  (X: nRows x 128 row-major, W: 128 x 128 row-major), fp32 WMMA.
// Block = 256 threads = 8 waves, covers 64 rows x 128 cols.
// Wave w owns N-tile w (16 cols) and 4 M-tiles -> 4 x v8f accumulators,
// 128 x V_WMMA_F32_16X16X4_F32 per wave.
// W is staged once per block into padded LDS via gfx1250 async-to-LDS loads
// (ASYNCcnt path); B fragments then come from ds_load, A from global b64.
__global__ __launch_bounds__(256) void k_gemm_wmma(const float* __restrict__ X,
                                                   const float* __restrict__ W,
                                                   float* __restrict__ H,
                                                   int nRows) {
  __shared__ float wlds[EMB * WSTRIDE];   // 128 x 132 floats = 67.6 KB

  const int tid  = threadIdx.x;
  const int lane = tid & 31;
  const int wv   = tid >> 5;            // 0..7 -> 16-col tile
  const int l15  = lane & 15;
  const int half = lane >> 4;           // 0|1 -> K-pair select
  const int col  = wv * 16 + l15;
  const int mBase = blockIdx.x * 64;

  // ---- stage W (128x128 f32) into LDS with async DMA loads --------------
  // 4096 float4 total; each thread issues 16 async b128 transfers.
  {
#pragma unroll
    for (int it = 0; it < 16; ++it) {
      const int idx4 = it * 256 + tid;          // float4 index in W
      const int row  = idx4 >> 5;               // /32 float4 per row
      const int c4   = idx4 & 31;
      unsigned lds_off =
          (unsigned)(unsigned long long)&wlds[row * WSTRIDE + c4 * 4];
      unsigned long long ga =
          (unsigned long long)(const void*)(W + (size_t)row * EMB + c4 * 4);
      asm volatile("global_load_async_to_lds_b128 %0, %1, off"
                   :: "v"(lds_off), "v"(ga) : "memory");
    }
    asm volatile("s_wait_asynccnt 0x0" ::: "memory");
    __syncthreads();
  }

  // ---- A-fragment row pointers (clamped: EXEC stays all-ones for WMMA) ---
  const float* xp0; const float* xp1; const float* xp2; const float* xp3;
  {
    int r0 = mBase + 0 * 16 + l15; if (r0 >= nRows) r0 = nRows - 1;
    int r1 = mBase + 1 * 16 + l15; if (r1 >= nRows) r1 = nRows - 1;
    int r2 = mBase + 2 * 16 + l15; if (r2 >= nRows) r2 = nRows - 1;
    int r3 = mBase + 3 * 16 + l15; if (r3 >= nRows) r3 = nRows - 1;
    xp0 = X + (size_t)r0 * EMB + 2 * half;
    xp1 = X + (size_t)r1 * EMB + 2 * half;
    xp2 = X + (size_t)r2 * EMB + 2 * half;
    xp3 = X + (size_t)r3 * EMB + 2 * half;
  }

  v8f acc0 = {}, acc1 = {}, acc2 = {}, acc3 = {};
#pragma unroll 4
  for (int k = 0; k < EMB; k += 4) {
    const v2f a0 = *(const v2f*)(xp0 + k);
    const v2f a1 = *(const v2f*)(xp1 + k);
    const v2f a2 = *(const v2f*)(xp2 + k);
    const v2f a3 = *(const v2f*)(xp3 + k);
    const int kk = k + 2 * half;
    v2f b;
    b.x = wlds[kk * WSTRIDE + col];
    b.y = wlds[(kk + 1) * WSTRIDE + col];
    acc0 = __builtin_amdgcn_wmma_f32_16x16x4_f32(false, a0, false, b, (short)0, acc0, false, false);
    acc1 = __builtin_amdgcn_wmma_f32_16x16x4_f32(false, a1, false, b, (short)0, acc1, false, false);
    acc2 = __builtin_amdgcn_wmma_f32_16x16x4_f32(false, a2, false, b, (short)0, acc2, false, false);
    acc3 = __builtin_amdgcn_wmma_f32_16x16x4_f32(false, a3, false, b, (short)0, acc3, false, false);
  }

  // ---- store: unguarded clause for full tiles (wave-uniform branch) ------
  const v8f accs[4] = {acc0, acc1, acc2, acc3};
#pragma unroll
  for (int t = 0; t < 4; ++t) {
    const int mt = mBase + t * 16;
    if (mt + 16 <= nRows) {
      float* hp = H + (size_t)(mt + 8 * half) * EMB + col;
#pragma unroll
      for (int v = 0; v < 8; ++v) hp[(size_t)v * EMB] = accs[t][v];
    } else if (mt < nRows) {              // partial tile (nRows % 16 != 0 only)
#pragma unroll
      for (int v = 0; v < 8; ++v) {
        const int m = mt + 8 * half + v;
        if (m < nRows) H[(size_t)m * EMB + col] = accs[t][v];
      }
    }
  }
}

// OUT = dinv^2 * H + bias  (self-loop contribution; initializes OUT so the
// scatter kernel can pure-atomically accumulate on top). float4-vectorized.
__global__ __launch_bounds__(256) void k_selfloop_bias(const float4* __restrict__ H4,
                                                       const float* __restrict__ dinv,
                                                       const float4* __restrict__ bias4,
                                                       float4* __restrict__ OUT4,
                                                       int total4) {
  int i = blockIdx.x * 256 + threadIdx.x;
  if (i >= total4) return;
  const int node = i >> 5;            // 32 float4 per node
  const int c4   = i & 31;
  const float di = dinv[node];
  const float s  = di * di;
  const float4 h = H4[i];
  const float4 b = bias4[c4];
  OUT4[i] = make_float4(s * h.x + b.x, s * h.y + b.y,
                        s * h.z + b.z, s * h.w + b.w);
}

// One wave32 per edge: lane loads float4 of h[src] (global_load_b128),
// scales by dinv[src]*dinv[dst], 4 native f32 atomics onto out[dst].
// h (51 MB) is L2-resident, so gathers + atomics stay in the 192 MB L2.
__global__ __launch_bounds__(256) void k_scatter(const float* __restrict__ H,
                                                 const float* __restrict__ dinv,
                                                 const int* __restrict__ srcA,
                                                 const int* __restrict__ dstA,
                                                 const int* __restrict__ srcB,
                                                 const int* __restrict__ dstB,
                                                 int E1, int totalE,
                                                 float* __restrict__ OUT) {
  const int gtid = blockIdx.x * 256 + threadIdx.x;
  const int e    = gtid >> 5;
  const int lane = gtid & 31;
  if (e >= totalE) return;

  int s, d;
  if (e < E1) { s = srcA[e]; d = dstA[e]; }
  else        { s = srcB[e - E1]; d = dstB[e - E1]; }

  const float norm = dinv[s] * dinv[d];
  const float4 v = ((const float4*)(H + (size_t)s * EMB))[lane];

  float* op = OUT + (size_t)d * EMB + lane * 4;
  atomic_fadd(op + 0, norm * v.x);
  atomic_fadd(op + 1, norm * v.y);
  atomic_fadd(op + 2, norm * v.z);
  atomic_fadd(op + 3, norm * v.w);
}

extern "C" void kernel_launch(void* const* d_in, const int* in_sizes, int n_in,
                              void* d_out, int out_size, void* d_ws, size_t ws_size,
                              hipStream_t stream) {
  const float* x   = (const float*)d_in[0];
  const float* W0  = (const float*)d_in[1];
  const float* b0  = (const float*)d_in[2];
  const float* W1  = (const float*)d_in[3];
  const float* b1  = (const float*)d_in[4];
  const int*  posE = (const int*)d_in[5];   // (2,E) flat: row0=src, row1=dst
  const int*  negE = (const int*)d_in[6];

  const int N      = in_sizes[0] / EMB;
  const int E1     = in_sizes[5] / 2;
  const int E2     = in_sizes[6] / 2;
  const int totalE = E1 + E2;

  const int* srcA = posE;       const int* dstA = posE + E1;
  const int* srcB = negE;       const int* dstB = negE + E2;

  // Workspace layout (floats): [deg:N int][dinv:N][h:N*128][z1:N*128] ~103 MB
  int*   deg  = (int*)d_ws;
  float* dinv = (float*)d_ws + N;
  float* h    = (float*)d_ws + 2 * (size_t)N;
  float* z1   = h + (size_t)N * EMB;
  float* out  = (float*)d_out;

  const int B = 256;
  const dim3 blk(B);
  const int    mBlocks = (N + 63) / 64;
  const int    total4  = N * (EMB / 4);
  const size_t scatThr = (size_t)totalE * 32;
  const int    scatBlk = (int)((scatThr + B - 1) / B);

  // Normalization (graph identical for both layers)
  k_zero_i32<<<(N + B - 1) / B, blk, 0, stream>>>(deg, N);
  k_degree  <<<(totalE + B - 1) / B, blk, 0, stream>>>(dstA, dstB, E1, totalE, deg);
  k_dinv    <<<(N + B - 1) / B, blk, 0, stream>>>(deg, dinv, N);

  // Layer 1: h = x@W0 ; z1 = dinv^2*h + b0 ; z1 += scatter(norm * h[src] -> dst)
  k_gemm_wmma    <<<mBlocks, blk, 0, stream>>>(x, W0, h, N);
  k_selfloop_bias<<<(total4 + B - 1) / B, blk, 0, stream>>>(
      (const float4*)h, dinv, (const float4*)b0, (float4*)z1, total4);
  k_scatter      <<<scatBlk, blk, 0, stream>>>(h, dinv, srcA, dstA, srcB, dstB,
                                               E1, totalE, z1);

  // Layer 2: h = z1@W1 ; out = dinv^2*h + b1 ; out += scatter
  k_gemm_wmma    <<<mBlocks, blk, 0, stream>>>(z1, W1, h, N);
  k_selfloop_bias<<<(total4 + B - 1) / B, blk, 0, stream>>>(
      (const float4*)h, dinv, (const float4*)b1, (float4*)out, total4);
  k_scatter      <<<scatBlk, blk, 0, stream>>>(h, dinv, srcA, dstA, srcB, dstB,
                                               E1, totalE, out);
}